// ECELoss_33638183862595
// MI455X (gfx1250) — compile-verified
//
#include <hip/hip_runtime.h>
#include <math.h>

// ----------------------------------------------------------------------------
// ECE loss, MI455X (gfx1250, wave32).
// Bandwidth-bound: 512MB logits @ 23.3TB/s => ~22us floor. No matrix math
// applies (argmax + softmax-denominator reduction); optimize the memory path:
// one wave per 128-float row, one global_load_b128 (non-temporal) per lane,
// wave32 max butterfly + ballot-based first-occurrence argmax, ds_add_f32
// block aggregation, 64-way replicated global atomic tables.
// ----------------------------------------------------------------------------

#define NBINS 15
#define NCOLS 128
#define WAVES_PER_BLOCK 8
#define BLOCK_THREADS (WAVES_PER_BLOCK * 32)
#define ROWS_PER_WAVE 16
#define ROWS_PER_BLOCK (WAVES_PER_BLOCK * ROWS_PER_WAVE)
#define NREP 64            // replicated global accumulator tables
#define REP_STRIDE 48      // 45 used (3*15), padded to 48 for alignment

typedef float v4f __attribute__((ext_vector_type(4)));

__global__ void ece_zero_kernel(float* __restrict__ tbl, int n) {
    int i = blockIdx.x * blockDim.x + threadIdx.x;
    if (i < n) tbl[i] = 0.0f;
}

__global__ __launch_bounds__(BLOCK_THREADS)
void ece_partial_kernel(const float* __restrict__ logits,
                        const int*   __restrict__ labels,
                        float*       __restrict__ tbl,
                        long long nrows) {
    __shared__ float sb[3 * NBINS];
    const int tid  = threadIdx.x;
    const int lane = tid & 31;
    const int wave = tid >> 5;

    if (tid < 3 * NBINS) sb[tid] = 0.0f;
    __syncthreads();

    const v4f* __restrict__ rp = (const v4f*)logits;   // NCOLS/4 = 32 v4f per row
    long long row = (long long)blockIdx.x * ROWS_PER_BLOCK + wave;

    // software pipeline: preload first row
    v4f v = (row < nrows) ? __builtin_nontemporal_load(&rp[(size_t)row * (NCOLS / 4) + lane])
                          : (v4f)(0.0f);

    for (int r = 0; r < ROWS_PER_WAVE; ++r) {
        const long long nrow = row + WAVES_PER_BLOCK;
        v4f vn = (v4f)(0.0f);
        if (r + 1 < ROWS_PER_WAVE && nrow < nrows) {
            vn = __builtin_nontemporal_load(&rp[(size_t)nrow * (NCOLS / 4) + lane]);
            // hint the row after that into GL2 (global_prefetch_b8)
            if (r + 2 < ROWS_PER_WAVE && nrow + WAVES_PER_BLOCK < nrows)
                __builtin_prefetch(&rp[(size_t)(nrow + WAVES_PER_BLOCK) * (NCOLS / 4) + lane], 0, 1);
        }

        if (row < nrows) {
            // ---- lane-local max of 4 ----
            float m = fmaxf(fmaxf(v.x, v.y), fmaxf(v.z, v.w));

            // ---- wave32 max butterfly (5 shuffles) ----
            #pragma unroll
            for (int off = 16; off > 0; off >>= 1)
                m = fmaxf(m, __shfl_xor(m, off, 32));

            // ---- first-occurrence argmax via ballot ----
            // lane l owns indices 4l..4l+3, so the lowest lane holding the max
            // owns the lowest index range; within it, take the first component.
            const bool has = (v.x == m) || (v.y == m) || (v.z == m) || (v.w == m);
            const unsigned long long bal = __ballot(has);
            const int firstlane = __ffsll((long long)bal) - 1;
            const int li = (v.x == m) ? 0 : ((v.y == m) ? 1 : ((v.z == m) ? 2 : 3));
            const int gidx = firstlane * 4 + __shfl(li, firstlane, 32);

            // ---- sum of exp(x - max); conf = exp(0)/sum = 1/sum ----
            float s = __expf(v.x - m) + __expf(v.y - m) +
                      __expf(v.z - m) + __expf(v.w - m);
            #pragma unroll
            for (int off = 16; off > 0; off >>= 1)
                s += __shfl_xor(s, off, 32);

            if (lane == 0) {
                const float conf = 1.0f / s;
                int b = (int)ceilf(conf * (float)NBINS) - 1;
                b = b < 0 ? 0 : (b > NBINS - 1 ? NBINS - 1 : b);
                const float acc = (gidx == labels[row]) ? 1.0f : 0.0f;
                atomicAdd(&sb[b],             1.0f);   // ds_add_f32
                atomicAdd(&sb[NBINS + b],     conf);
                atomicAdd(&sb[2 * NBINS + b], acc);
            }
        }

        v = vn;
        row = nrow;
    }

    __syncthreads();
    if (tid < 3 * NBINS) {
        // replicate across NREP tables: contention per address ~ blocks/NREP
        float* dst = tbl + (size_t)(blockIdx.x & (NREP - 1)) * REP_STRIDE + tid;
        atomicAdd(dst, sb[tid]);                       // global_atomic_add_f32
    }
}

__global__ void ece_final_kernel(const float* __restrict__ tbl,
                                 float* __restrict__ out, float n) {
    __shared__ float merged[3 * NBINS];
    const int t = threadIdx.x;
    if (t < 3 * NBINS) {
        float s = 0.0f;
        for (int r = 0; r < NREP; ++r) s += tbl[(size_t)r * REP_STRIDE + t];
        merged[t] = s;
    }
    __syncthreads();
    if (t == 0) {
        float ece = 0.0f;
        #pragma unroll
        for (int b = 0; b < NBINS; ++b) {
            const float cnt  = merged[b];
            const float cs   = merged[NBINS + b];
            const float as   = merged[2 * NBINS + b];
            const float safe = fmaxf(cnt, 1.0f);
            const float pb   = fabsf(cs / safe - as / safe) * (cnt / n);
            if (cnt > 0.0f) ece += pb;
        }
        out[0] = ece;
    }
}

extern "C" void kernel_launch(void* const* d_in, const int* in_sizes, int n_in,
                              void* d_out, int out_size, void* d_ws, size_t ws_size,
                              hipStream_t stream) {
    const float* logits = (const float*)d_in[0];
    const int*   labels = (const int*)d_in[1];      // int32 per harness convention
    float* out = (float*)d_out;
    float* tbl = (float*)d_ws;

    const long long N = (long long)in_sizes[1];     // label count == row count

    const int tbl_elems = NREP * REP_STRIDE;
    ece_zero_kernel<<<(tbl_elems + 255) / 256, 256, 0, stream>>>(tbl, tbl_elems);

    const long long nblocks = (N + ROWS_PER_BLOCK - 1) / ROWS_PER_BLOCK;
    ece_partial_kernel<<<(int)nblocks, BLOCK_THREADS, 0, stream>>>(logits, labels, tbl, N);

    ece_final_kernel<<<1, 64, 0, stream>>>(tbl, out, (float)N);
}